// CNN_FC_GraphSAGE_4045859193433
// MI455X (gfx1250) — compile-verified
//
#include <hip/hip_runtime.h>
#include <hip/hip_bf16.h>

typedef __attribute__((ext_vector_type(2))) float v2f;
typedef __attribute__((ext_vector_type(8))) float v8f;

#define NNODES 20000
#define NBATCH 16
#define NEDGE  320000
#define NROWS  320000   // NNODES * NBATCH

// ---------------------------------------------------------------------------
// Kernel 1: per-node MLP front-end. One wave32 block handles the 16 rows
// (= all 16 batches) of one node: conv branches -> feat[16x32] -> fc1 via
// v_wmma_f32_16x16x4_f32 -> ReLU -> fc2 via WMMA -> h0[row][5].
// ---------------------------------------------------------------------------
__global__ __launch_bounds__(32) void k_mlp(
    const float* __restrict__ x,
    const float* __restrict__ w_t, const float* __restrict__ b_t,
    const float* __restrict__ w_r, const float* __restrict__ b_r,
    const float* __restrict__ w_tp, const float* __restrict__ b_tp,
    const float* __restrict__ w_ssr, const float* __restrict__ b_ssr,
    const float* __restrict__ fc1_w, const float* __restrict__ fc1_b,
    const float* __restrict__ fc2_w, const float* __restrict__ fc2_b,
    float* __restrict__ h0)
{
    __shared__ __align__(16) float xs_s[16][108];   // 16 rows of raw features
    __shared__ float feat_s[16][32];                // concat(passthrough, branches)
    __shared__ float h1_s[16][20];                  // fc1 output staging
    __shared__ float wb_s[480];                     // 4 branches x 5 ch x 24 taps
    __shared__ float bb_s[20];                      // 4 x 5 biases

    const int lane = threadIdx.x;      // 0..31 (wave32)
    const int node = blockIdx.x;       // 0..19999
    const int half = lane >> 4;        // 0: K0,K1 ; 1: K2,K3 (A/B frag halves)
    const int lm   = lane & 15;

    // --- preload branch weights to LDS ---
    {
        const float* wsrc[4] = {w_t, w_r, w_tp, w_ssr};
        for (int j = lane; j < 480; j += 32) wb_s[j] = wsrc[j / 120][j % 120];
        const float* bsrc[4] = {b_t, b_r, b_tp, b_ssr};
        if (lane < 20) bb_s[lane] = bsrc[lane / 5][lane % 5];
    }

    // --- cooperative vectorized load of 16 rows x 108 floats (432B/row) ---
    for (int j = lane; j < 16 * 27; j += 32) {
        int m = j / 27, q = j % 27;  // m = batch index = row-in-tile
        const float4* src = (const float4*)(x + ((size_t)m * NNODES + node) * 108);
        ((float4*)xs_s[m])[q] = src[q];
    }
    __syncthreads();

    // --- conv branches (length-24 dots, 5 ch x 4 branches) + ReLU ---
    for (int o = lane; o < 16 * 20; o += 32) {
        int m = o / 20, c = o % 20;
        int br = c / 5, ch = c % 5;
        const float* wrow = &wb_s[br * 120 + ch * 24];
        const float* xrow = &xs_s[m][12 + br * 24];
        float s = bb_s[c];
        #pragma unroll
        for (int k = 0; k < 24; k++) s += xrow[k] * wrow[k];
        feat_s[m][12 + c] = fmaxf(s, 0.0f);
    }
    // --- passthrough first 12 features ---
    for (int j = lane; j < 16 * 12; j += 32) {
        int m = j / 12, c = j % 12;
        feat_s[m][c] = xs_s[m][c];
    }
    __syncthreads();

    // --- fc1: [16x32] x [32x20] via f32 WMMA, two N-tiles (0..15, 16..19) ---
    v8f acc0 = {};
    v8f acc1 = {};
    #pragma unroll
    for (int kc = 0; kc < 32; kc += 4) {
        int k0 = kc + 2 * half;
        v2f a; a.x = feat_s[lm][k0]; a.y = feat_s[lm][k0 + 1];
        // B tile 0: N = lm (all valid, 20>16)
        v2f b0; b0.x = fc1_w[lm * 32 + k0]; b0.y = fc1_w[lm * 32 + k0 + 1];
        // B tile 1: N = 16+lm, valid only for lm<4 (mask, no divergence)
        int n1 = 16 + lm;
        float msk = (n1 < 20) ? 1.0f : 0.0f;
        int n1c = (n1 < 20) ? n1 : 0;
        v2f b1; b1.x = fc1_w[n1c * 32 + k0] * msk; b1.y = fc1_w[n1c * 32 + k0 + 1] * msk;
        acc0 = __builtin_amdgcn_wmma_f32_16x16x4_f32(false, a, false, b0, (short)0, acc0, false, false);
        acc1 = __builtin_amdgcn_wmma_f32_16x16x4_f32(false, a, false, b1, (short)0, acc1, false, false);
    }
    // bias + ReLU, stage h1 to LDS.  C/D layout: VGPR g -> M = g + 8*half, N = lm(+16)
    {
        float bias0 = fc1_b[lm];
        #pragma unroll
        for (int g = 0; g < 8; g++)
            h1_s[g + 8 * half][lm] = fmaxf(acc0[g] + bias0, 0.0f);
        if (lm < 4) {
            float bias1 = fc1_b[16 + lm];
            #pragma unroll
            for (int g = 0; g < 8; g++)
                h1_s[g + 8 * half][16 + lm] = fmaxf(acc1[g] + bias1, 0.0f);
        }
    }
    __syncthreads();

    // --- fc2: [16x20] x [20x5] via f32 WMMA (single N-tile, cols 5..15 zero) ---
    v8f acc2v = {};
    float msk2 = (lm < 5) ? 1.0f : 0.0f;
    int n2 = (lm < 5) ? lm : 0;
    #pragma unroll
    for (int kc = 0; kc < 20; kc += 4) {
        int k0 = kc + 2 * half;
        v2f a; a.x = h1_s[lm][k0]; a.y = h1_s[lm][k0 + 1];
        v2f b; b.x = fc2_w[n2 * 20 + k0] * msk2; b.y = fc2_w[n2 * 20 + k0 + 1] * msk2;
        acc2v = __builtin_amdgcn_wmma_f32_16x16x4_f32(false, a, false, b, (short)0, acc2v, false, false);
    }
    // bias + store h0[row][c], row = node*16 + m  (m = batch)
    if (lm < 5) {
        float bo = fc2_b[lm];
        #pragma unroll
        for (int g = 0; g < 8; g++) {
            int m = g + 8 * half;
            h0[((size_t)node * 16 + m) * 5 + lm] = acc2v[g] + bo;
        }
    }
}

// ---------------------------------------------------------------------------
// Kernel 2: SAGE1 edge scatter. deg[dst]++ and agg1[dst][b][c] += h0[src][b][c]
// ---------------------------------------------------------------------------
__global__ void k_scatter1(const int* __restrict__ ei,
                           const float* __restrict__ h0,
                           float* __restrict__ agg1,
                           float* __restrict__ deg)
{
    int e = blockIdx.x * blockDim.x + threadIdx.x;
    if (e >= NEDGE) return;
    int s = ei[e];
    int d = ei[NEDGE + e];
    atomicAdd(&deg[d], 1.0f);
    const float* hs = h0 + (size_t)s * 80;   // 16 batches x 5 ch, contiguous
    float* ad = agg1 + (size_t)d * 80;
    #pragma unroll 8
    for (int j = 0; j < 80; j++) atomicAdd(&ad[j], hs[j]);
}

// ---------------------------------------------------------------------------
// Kernel 3: SAGE1 node update fused with SAGE2 projections.
// h1 (20 dims) stays in registers; reduce to s = g2_wl.h1, r = g2_wr.h1.
// ---------------------------------------------------------------------------
__global__ void k_node1(const float* __restrict__ h0,
                        const float* __restrict__ agg1,
                        const float* __restrict__ deg,
                        const float* __restrict__ g1_wl,
                        const float* __restrict__ g1_bl,
                        const float* __restrict__ g1_wr,
                        const float* __restrict__ g2_wl,
                        const float* __restrict__ g2_wr,
                        float* __restrict__ sbuf,
                        float* __restrict__ rbuf)
{
    int i = blockIdx.x * blockDim.x + threadIdx.x;   // i = node*16 + batch
    if (i >= NROWS) return;
    int n = i >> 4;
    float inv = 1.0f / fmaxf(deg[n], 1.0f);
    float mean[5], root[5];
    #pragma unroll
    for (int c = 0; c < 5; c++) {
        mean[c] = agg1[(size_t)i * 5 + c] * inv;
        root[c] = h0[(size_t)i * 5 + c];
    }
    float sv = 0.0f, rv = 0.0f;
    #pragma unroll
    for (int j = 0; j < 20; j++) {
        float h = g1_bl[j];
        #pragma unroll
        for (int c = 0; c < 5; c++)
            h += g1_wl[j * 5 + c] * mean[c] + g1_wr[j * 5 + c] * root[c];
        sv += g2_wl[j] * h;
        rv += g2_wr[j] * h;
    }
    sbuf[i] = sv;
    rbuf[i] = rv;
}

// ---------------------------------------------------------------------------
// Kernel 4: SAGE2 scalar edge scatter. acc2[dst][b] += s[src][b]
// ---------------------------------------------------------------------------
__global__ void k_scatter2(const int* __restrict__ ei,
                           const float* __restrict__ sbuf,
                           float* __restrict__ acc2)
{
    int e = blockIdx.x * blockDim.x + threadIdx.x;
    if (e >= NEDGE) return;
    int s = ei[e];
    int d = ei[NEDGE + e];
    const float* ss = sbuf + (size_t)s * 16;
    float* ad = acc2 + (size_t)d * 16;
    #pragma unroll
    for (int b = 0; b < 16; b++) atomicAdd(&ad[b], ss[b]);
}

// ---------------------------------------------------------------------------
// Kernel 5: final combine, out[b*N + n] = acc2/deg + bias + root
// ---------------------------------------------------------------------------
__global__ void k_final(const float* __restrict__ acc2,
                        const float* __restrict__ rbuf,
                        const float* __restrict__ deg,
                        const float* __restrict__ g2_bl,
                        float* __restrict__ out)
{
    int i = blockIdx.x * blockDim.x + threadIdx.x;   // i = node*16 + batch
    if (i >= NROWS) return;
    int n = i >> 4, b = i & 15;
    float inv = 1.0f / fmaxf(deg[n], 1.0f);
    out[(size_t)b * NNODES + n] = acc2[i] * inv + g2_bl[0] + rbuf[i];
}

// ---------------------------------------------------------------------------
extern "C" void kernel_launch(void* const* d_in, const int* in_sizes, int n_in,
                              void* d_out, int out_size, void* d_ws, size_t ws_size,
                              hipStream_t stream) {
    const float* x     = (const float*)d_in[0];
    const int*   ei    = (const int*)  d_in[1];
    const float* w_t   = (const float*)d_in[2];
    const float* b_t   = (const float*)d_in[3];
    const float* w_r   = (const float*)d_in[4];
    const float* b_r   = (const float*)d_in[5];
    const float* w_tp  = (const float*)d_in[6];
    const float* b_tp  = (const float*)d_in[7];
    const float* w_ssr = (const float*)d_in[8];
    const float* b_ssr = (const float*)d_in[9];
    const float* fc1_w = (const float*)d_in[10];
    const float* fc1_b = (const float*)d_in[11];
    const float* fc2_w = (const float*)d_in[12];
    const float* fc2_b = (const float*)d_in[13];
    const float* g1_wl = (const float*)d_in[14];
    const float* g1_bl = (const float*)d_in[15];
    const float* g1_wr = (const float*)d_in[16];
    const float* g2_wl = (const float*)d_in[17];
    const float* g2_bl = (const float*)d_in[18];
    const float* g2_wr = (const float*)d_in[19];
    float* out = (float*)d_out;

    // workspace layout (~16.7 MB of floats)
    float* ws   = (float*)d_ws;
    float* h0   = ws;                         // NROWS * 5
    float* agg1 = h0 + (size_t)NROWS * 5;     // NROWS * 5
    float* deg  = agg1 + (size_t)NROWS * 5;   // NNODES
    float* sbuf = deg + NNODES;               // NROWS
    float* rbuf = sbuf + NROWS;               // NROWS
    float* acc2 = rbuf + NROWS;               // NROWS

    hipMemsetAsync(agg1, 0, (size_t)NROWS * 5 * sizeof(float), stream);
    hipMemsetAsync(deg,  0, (size_t)NNODES * sizeof(float), stream);
    hipMemsetAsync(acc2, 0, (size_t)NROWS * sizeof(float), stream);

    k_mlp<<<NNODES, 32, 0, stream>>>(x, w_t, b_t, w_r, b_r, w_tp, b_tp, w_ssr, b_ssr,
                                     fc1_w, fc1_b, fc2_w, fc2_b, h0);
    k_scatter1<<<(NEDGE + 255) / 256, 256, 0, stream>>>(ei, h0, agg1, deg);
    k_node1<<<(NROWS + 255) / 256, 256, 0, stream>>>(h0, agg1, deg,
                                                     g1_wl, g1_bl, g1_wr,
                                                     g2_wl, g2_wr, sbuf, rbuf);
    k_scatter2<<<(NEDGE + 255) / 256, 256, 0, stream>>>(ei, sbuf, acc2);
    k_final<<<(NROWS + 255) / 256, 256, 0, stream>>>(acc2, rbuf, deg, g2_bl, out);
}